// KernelWarehouseDynConv_6863357739679
// MI455X (gfx1250) — compile-verified
//
#include <hip/hip_runtime.h>
#include <hip/hip_bf16.h>

// Problem constants (match the JAX reference)
#define BB 16
#define CC 64
#define HH 256
#define WW 256
#define KK 4

#define TILE_W 64            // pixels per workgroup (one row strip)
#define BLOCK_T 128          // 4 wave32s, each covers 16 pixels
#define LDS_COLS (TILE_W + 2)
#define NCHUNK 144           // 9 taps * (64/4) channel chunks

// LDS layout: [cpair 0..31][row 0..2][col 0..65][2 channels] as b64 elements,
// each cpair plane padded to 416 dwords (== 32 mod 64) so the two lane-halves
// of a wave32 b64 read hit disjoint bank sets (0-31 vs 32-63).
#define PLANE_F  416                 // floats per cpair plane (3*66*2=396 + 20 pad)
#define PLANE_V2 (PLANE_F / 2)       // 208 v2f elements per cpair plane
#define LDS_TOT  (32 * PLANE_F)      // 13312 floats = 53,248 bytes

typedef __attribute__((ext_vector_type(2))) float v2f;
typedef __attribute__((ext_vector_type(8))) float v8f;

// ---------------------------------------------------------------------------
// Prep kernel: pack A-matrix fragments (conv weights rows 0-3, attn 1x1
// weights as center-tap rows 4-7, rows 8-15 zero) into the per-lane VGPR
// layout of V_WMMA_F32_16X16X4_F32's A operand:
//   chunk q (= tap*16 + c0/4), lane l, vgpr j -> Apack[q*64 + l*2 + j]
// 144*64 floats = 36,864 B in d_ws; identical for every block, L2-resident.
// ---------------------------------------------------------------------------
__global__ __launch_bounds__(256)
void kw_pack_a(const float* __restrict__ weight,   // [4,64,3,3]
               const float* __restrict__ attn_w,   // [4,64]
               float* __restrict__ Apack)
{
    int idx = blockIdx.x * blockDim.x + threadIdx.x;
    if (idx >= NCHUNK * 64) return;
    int q  = idx >> 6;            // chunk 0..143
    int e  = idx & 63;            // l*2 + j
    int l  = e >> 1;
    int j  = e & 1;
    int t  = q >> 4;              // tap 0..8
    int c0 = (q & 15) << 2;
    int m  = l & 15;
    int k  = ((l >> 4) << 1) + j;
    int c  = c0 + k;
    int dy = t / 3, dx = t % 3;
    float v = 0.0f;
    if (m < KK)                       v = weight[((m * CC + c) * 3 + dy) * 3 + dx];
    else if (m < 2 * KK && t == 4)    v = attn_w[(m - KK) * CC + c];   // center tap
    Apack[idx] = v;
}

// ---------------------------------------------------------------------------
// Fused main kernel: stage x tile in LDS once, run the 16x16x576 implicit-GEMM
// with fp32 WMMA producing conv[0..3] and logits[0..3] per pixel, then
// softmax + mixture epilogue. Inner loop per step:
//   1x global_load_b64 (A, SGPR base + imm, L2-hot)
//   1x ds_load_b64     (B, single base VGPR + imm offset, conflict-free)
//   1x v_wmma_f32_16x16x4_f32
// ---------------------------------------------------------------------------
__global__ __launch_bounds__(BLOCK_T)
void kw_dynconv_fused(const float* __restrict__ x,
                      const float* __restrict__ Apack,    // [144*64] packed A
                      const float* __restrict__ attn_b,   // [4]
                      float* __restrict__ out)            // [16,1,256,256]
{
    __shared__ float xsf[LDS_TOT];   // 53,248 B

    const int blk   = blockIdx.x;
    const int wtile = blk & ((WW / TILE_W) - 1);          // 0..3
    const int tmp   = blk >> 2;                           // /(W/TILE_W)
    const int h     = tmp & (HH - 1);
    const int b     = tmp >> 8;                           // /H
    const int w0    = wtile * TILE_W;

    const int tid = threadIdx.x;

    // ---- Stage the x tile (SAME-padding zeros) into the padded pair layout --
    const int TOT = CC * 3 * LDS_COLS;                    // 12672 = 99 * 128
    for (int i = tid; i < TOT; i += BLOCK_T) {
        int col = i % LDS_COLS;
        int r   = (i / LDS_COLS) % 3;
        int c   = i / (3 * LDS_COLS);
        int gh  = h + r - 1;
        int gw  = w0 + col - 1;
        float v = 0.0f;
        if ((unsigned)gh < (unsigned)HH && (unsigned)gw < (unsigned)WW)
            v = x[(((b * CC) + c) * HH + gh) * WW + gw];
        xsf[(c >> 1) * PLANE_F + (r * LDS_COLS + col) * 2 + (c & 1)] = v;
    }
    __syncthreads();

    const int wave = tid >> 5;        // 0..3
    const int lane = tid & 31;
    const int n    = lane & 15;       // N column (pixel)
    const int hi   = lane >> 4;       // lane half -> B rows k = hi*2, hi*2+1
    const int wcol = wave * 16 + n;   // pixel column within tile

    // Single per-lane bases; all loop offsets become immediates after unroll.
    const float* __restrict__ Ap = Apack + (lane << 1);
    const v2f*  __restrict__ xv  = reinterpret_cast<const v2f*>(xsf)
                                   + hi * PLANE_V2 + wcol;

    // Accumulator: rows 0-3 = conv[k], rows 4-7 = logits[k], 8-15 = zero
    v8f acc = {};

#pragma unroll
    for (int t = 0; t < 9; ++t) {
        const int dy = t / 3;
        const int dx = t % 3;
#pragma unroll
        for (int cq = 0; cq < 16; ++cq) {
            const int q = t * 16 + cq;
            v2f a  = *reinterpret_cast<const v2f*>(Ap + q * 64);
            // B[k = hi*2 + j][n], j=0,1: one aligned b64 element
            v2f bm = xv[(cq * 2) * PLANE_V2 + dy * LDS_COLS + dx];
            acc = __builtin_amdgcn_wmma_f32_16x16x4_f32(
                false, a, false, bm, (short)0, acc, false, false);
        }
    }

    // ---- Epilogue: lanes 0-15 own rows 0-7 for pixel n ----
    if (hi == 0) {
        float l0 = acc[4] + attn_b[0];
        float l1 = acc[5] + attn_b[1];
        float l2 = acc[6] + attn_b[2];
        float l3 = acc[7] + attn_b[3];
        float mx = fmaxf(fmaxf(l0, l1), fmaxf(l2, l3));
        float e0 = __expf(l0 - mx);
        float e1 = __expf(l1 - mx);
        float e2 = __expf(l2 - mx);
        float e3 = __expf(l3 - mx);
        float inv = 1.0f / (e0 + e1 + e2 + e3);
        float r = (acc[0] * e0 + acc[1] * e1 + acc[2] * e2 + acc[3] * e3) * inv;
        out[(b * HH + h) * WW + (w0 + wave * 16 + n)] = r;
    }
}

extern "C" void kernel_launch(void* const* d_in, const int* in_sizes, int n_in,
                              void* d_out, int out_size, void* d_ws, size_t ws_size,
                              hipStream_t stream) {
    (void)in_sizes; (void)n_in; (void)out_size; (void)ws_size;
    const float* x      = (const float*)d_in[0];
    const float* weight = (const float*)d_in[1];
    const float* attn_w = (const float*)d_in[2];
    const float* attn_b = (const float*)d_in[3];
    float* out   = (float*)d_out;
    float* Apack = (float*)d_ws;                  // 144*64*4 = 36,864 bytes

    // 1) Pack A fragments (same every call -> deterministic)
    kw_pack_a<<<(NCHUNK * 64 + 255) / 256, 256, 0, stream>>>(weight, attn_w, Apack);

    // 2) Fused conv + attention kernel
    const int grid = BB * HH * (WW / TILE_W);     // 16384 blocks
    kw_dynconv_fused<<<grid, BLOCK_T, 0, stream>>>(x, Apack, attn_b, out);
}